// GATConv_71038759076295
// MI455X (gfx1250) — compile-verified
//
#include <hip/hip_runtime.h>
#include <hip/hip_bf16.h>
#include <math.h>

#define N_NODES 10000
#define IN_FEAT 128
#define OUT_FEAT 64
#define N_EDGES 320000
#define LRELU_ALPHA 0.2f

typedef __attribute__((ext_vector_type(2))) float v2f;
typedef __attribute__((ext_vector_type(8))) float v8f;

// ---------------------------------------------------------------------------
// Init: zero d_out, rowsum, dedup bitmap; rowmax = -inf
// ---------------------------------------------------------------------------
__global__ void init_kernel(float* __restrict__ out, float* __restrict__ rowsum,
                            float* __restrict__ rowmax, unsigned int* __restrict__ bitmap,
                            int n_bitmap_words) {
    int t = blockIdx.x * blockDim.x + threadIdx.x;
    if (t < n_bitmap_words) bitmap[t] = 0u;
    if (t < N_NODES * OUT_FEAT) out[t] = 0.0f;
    if (t < N_NODES) {
        rowsum[t] = 0.0f;
        rowmax[t] = __uint_as_float(0xFF800000u);  // -inf
    }
}

// ---------------------------------------------------------------------------
// Wh = h @ W via V_WMMA_F32_16X16X4_F32 (fp32, matches reference precision).
// Block = 128 threads (4 waves). Block owns 16 rows x 64 cols of Wh.
// Wave w owns the 16-wide column tile [16w, 16w+16). K-loop: 32 WMMAs.
// A 16x4 layout (ISA 7.12.2): lane<16 -> K=0,1 ; lane>=16 -> K=2,3 ; M=lane&15.
// B 4x16 mirrored: N=lane&15, same K split. C/D: VGPR r holds M=r (+8 for
// lanes 16-31), N = lane&15.
// ---------------------------------------------------------------------------
__global__ __launch_bounds__(128) void wh_gemm_wmma(const float* __restrict__ h,
                                                    const float* __restrict__ W,
                                                    float* __restrict__ Wh) {
    const int lane    = threadIdx.x & 31;
    const int wave    = threadIdx.x >> 5;       // 0..3 -> column tile
    const int rowbase = blockIdx.x * 16;        // 625 blocks * 16 = 10000 exact
    const int colbase = wave * 16;
    const int m       = lane & 15;
    const int khalf   = (lane >> 4) * 2;        // 0 or 2

    const float* __restrict__ hrow = h + (size_t)(rowbase + m) * IN_FEAT + khalf;
    const float* __restrict__ wcol = W + colbase + m;   // W[k, col]

    v8f c = {};
#pragma unroll
    for (int k = 0; k < IN_FEAT; k += 4) {
        v2f a;
        a.x = hrow[k + 0];                       // h[row, k+khalf]   (b64 pair)
        a.y = hrow[k + 1];                       // h[row, k+khalf+1]
        v2f b;
        b.x = wcol[(size_t)(k + khalf + 0) * OUT_FEAT];
        b.y = wcol[(size_t)(k + khalf + 1) * OUT_FEAT];
        c = __builtin_amdgcn_wmma_f32_16x16x4_f32(false, a, false, b,
                                                  (short)0, c, false, false);
    }

    const int outcol = colbase + (lane & 15);
    const int rowoff = rowbase + ((lane >> 4) * 8);
#pragma unroll
    for (int r = 0; r < 8; ++r)
        Wh[(size_t)(rowoff + r) * OUT_FEAT + outcol] = c[r];
}

// ---------------------------------------------------------------------------
// Per-node attention scores: s1[i] = Wh[i]·a[0:64], s2[i] = Wh[i]·a[64:128]
// ---------------------------------------------------------------------------
__global__ void node_scores(const float* __restrict__ Wh, const float* __restrict__ a,
                            float* __restrict__ s1, float* __restrict__ s2) {
    int i = blockIdx.x * blockDim.x + threadIdx.x;
    if (i >= N_NODES) return;
    const float* w = Wh + (size_t)i * OUT_FEAT;
    float acc1 = 0.0f, acc2 = 0.0f;
#pragma unroll 8
    for (int f = 0; f < OUT_FEAT; ++f) {
        float v = w[f];
        acc1 += v * a[f];
        acc2 += v * a[OUT_FEAT + f];
    }
    s1[i] = acc1;
    s2[i] = acc2;
}

// Column means of Wh (fallback for rows with no edges: uniform softmax => mean)
__global__ void col_mean(const float* __restrict__ Wh, float* __restrict__ cm) {
    int f = threadIdx.x;  // 64 threads
    float s = 0.0f;
    for (int i = 0; i < N_NODES; ++i) s += Wh[(size_t)i * OUT_FEAT + f];
    cm[f] = s * (1.0f / (float)N_NODES);
}

__device__ __forceinline__ float lrelu(float v) {
    return v > 0.0f ? v : LRELU_ALPHA * v;
}

// float atomic max via signed/unsigned int monotone mapping (valid w/ -inf init)
__device__ __forceinline__ void atomicMaxFloat(float* addr, float val) {
    if (val >= 0.0f) atomicMax((int*)addr, __float_as_int(val));
    else             atomicMin((unsigned int*)addr, __float_as_uint(val));
}

// ---------------------------------------------------------------------------
// Pass A: per-row max of edge logits (duplicates harmless for max)
// ---------------------------------------------------------------------------
__global__ void edge_max(const long long* __restrict__ ei, const float* __restrict__ s1,
                         const float* __restrict__ s2, float* __restrict__ rowmax) {
    int t = blockIdx.x * blockDim.x + threadIdx.x;
    if (t >= N_EDGES) return;
    int i = (int)ei[t];             // src
    int j = (int)ei[N_EDGES + t];   // dst
    atomicMaxFloat(&rowmax[i], lrelu(s1[i] + s2[j]));
}

// ---------------------------------------------------------------------------
// Pass B: dedup (dense reference sets adj entry to 1 once) + sum of exp.
// First thread to set the (i,j) bit owns the edge; dups get weight 0.
// ---------------------------------------------------------------------------
__global__ void edge_sumexp(const long long* __restrict__ ei, const float* __restrict__ s1,
                            const float* __restrict__ s2, const float* __restrict__ rowmax,
                            unsigned int* __restrict__ bitmap, float* __restrict__ ews,
                            float* __restrict__ rowsum) {
    int t = blockIdx.x * blockDim.x + threadIdx.x;
    if (t >= N_EDGES) return;
    int i = (int)ei[t];
    int j = (int)ei[N_EDGES + t];
    long long key = (long long)i * N_NODES + j;
    unsigned int bit = 1u << (int)(key & 31);
    unsigned int old = atomicOr(&bitmap[key >> 5], bit);
    if (old & bit) { ews[t] = 0.0f; return; }   // duplicate edge
    float w = __expf(lrelu(s1[i] + s2[j]) - rowmax[i]);
    ews[t] = w;
    atomicAdd(&rowsum[i], w);
}

// ---------------------------------------------------------------------------
// Pass C: out[src] += (ews/rowsum[src]) * Wh[dst].  64 threads per edge.
// Wh is 2.5 MB -> L2-resident gathers.
// ---------------------------------------------------------------------------
__global__ void edge_accum(const long long* __restrict__ ei, const float* __restrict__ ews,
                           const float* __restrict__ rowsum, const float* __restrict__ Wh,
                           float* __restrict__ out) {
    long long t = (long long)blockIdx.x * blockDim.x + threadIdx.x;
    int e = (int)(t >> 6);
    int f = (int)(t & 63);
    if (e >= N_EDGES) return;
    float w = ews[e];
    if (w == 0.0f) return;                      // duplicate / no contribution
    int i = (int)ei[e];
    int j = (int)ei[N_EDGES + e];
    w /= rowsum[i];
    atomicAdd(&out[(size_t)i * OUT_FEAT + f], w * Wh[(size_t)j * OUT_FEAT + f]);
}

// ---------------------------------------------------------------------------
// Finalize: empty-row fallback (uniform softmax => column mean) then ELU
// ---------------------------------------------------------------------------
__global__ void finalize(float* __restrict__ out, const float* __restrict__ rowsum,
                         const float* __restrict__ cm) {
    int t = blockIdx.x * blockDim.x + threadIdx.x;
    if (t >= N_NODES * OUT_FEAT) return;
    int i = t >> 6;
    int f = t & 63;
    float x = (rowsum[i] > 0.0f) ? out[t] : cm[f];
    out[t] = x > 0.0f ? x : (__expf(x) - 1.0f);   // ELU
}

// ---------------------------------------------------------------------------
extern "C" void kernel_launch(void* const* d_in, const int* in_sizes, int n_in,
                              void* d_out, int out_size, void* d_ws, size_t ws_size,
                              hipStream_t stream) {
    const float*     h  = (const float*)d_in[0];
    const long long* ei = (const long long*)d_in[1];   // int64 [2, E]
    const float*     W  = (const float*)d_in[2];
    const float*     a  = (const float*)d_in[3];
    float* out = (float*)d_out;

    // workspace carve-up (256B aligned), total ~16.5 MB
    char* ws = (char*)d_ws;
    size_t off = 0;
    auto carve = [&](size_t bytes) {
        void* p = ws + off;
        off += (bytes + 255) & ~(size_t)255;
        return p;
    };
    const int bitmap_words = (N_NODES * N_NODES + 31) / 32;  // 3,125,000
    float*        Wh      = (float*)carve((size_t)N_NODES * OUT_FEAT * sizeof(float));
    float*        s1      = (float*)carve(N_NODES * sizeof(float));
    float*        s2      = (float*)carve(N_NODES * sizeof(float));
    float*        rowmax  = (float*)carve(N_NODES * sizeof(float));
    float*        rowsum  = (float*)carve(N_NODES * sizeof(float));
    float*        cm      = (float*)carve(OUT_FEAT * sizeof(float));
    float*        ews     = (float*)carve(N_EDGES * sizeof(float));
    unsigned int* bitmap  = (unsigned int*)carve((size_t)bitmap_words * sizeof(unsigned int));

    int init_elems = bitmap_words;  // largest of the init targets
    init_kernel<<<(init_elems + 255) / 256, 256, 0, stream>>>(out, rowsum, rowmax,
                                                              bitmap, bitmap_words);

    wh_gemm_wmma<<<N_NODES / 16, 128, 0, stream>>>(h, W, Wh);

    node_scores<<<(N_NODES + 255) / 256, 256, 0, stream>>>(Wh, a, s1, s2);
    col_mean<<<1, OUT_FEAT, 0, stream>>>(Wh, cm);

    edge_max<<<(N_EDGES + 255) / 256, 256, 0, stream>>>(ei, s1, s2, rowmax);
    edge_sumexp<<<(N_EDGES + 255) / 256, 256, 0, stream>>>(ei, s1, s2, rowmax,
                                                           bitmap, ews, rowsum);
    edge_accum<<<(N_EDGES * 64) / 256, 256, 0, stream>>>(ei, ews, rowsum, Wh, out);

    finalize<<<(N_NODES * OUT_FEAT + 255) / 256, 256, 0, stream>>>(out, rowsum, cm);
}